// SelfAttention_24970939859665
// MI455X (gfx1250) — compile-verified
//
#include <hip/hip_runtime.h>
#include <math.h>
#include <stdint.h>

// ---------------------------------------------------------------------------
// Multi-head self attention forward for MI455X (gfx1250).
//   fp32 WMMA (v_wmma_f32_16x16x4_f32) + async global->LDS staging pipeline.
//   E=1024, H=16, D=64, N=2, L=2048
// Outputs (concatenated, f32): out | q | k | v | energy | attention
// ---------------------------------------------------------------------------

#define EMBED 1024
#define HEADS 16
#define HDIM  64
#define NBAT  2
#define SEQ   2048
#define KC    64   // K-chunk staged in LDS: 64 x 64 f32 = 16 KB

typedef __attribute__((ext_vector_type(2))) float v2f;
typedef __attribute__((ext_vector_type(8))) float v8f;

// D(16x16,f32) = A(16x4,f32) x B(4x16,f32) + C
__device__ __forceinline__ v8f wmma4(v2f a, v2f b, v8f c) {
  return __builtin_amdgcn_wmma_f32_16x16x4_f32(
      false, a, false, b, (short)0, c, false, false);
}

// A fragment, row-major source, K walks along the row.
// ISA 32-bit A 16x4: lanes 0-15 -> K={0,1}, lanes 16-31 -> K={2,3}.
__device__ __forceinline__ v2f fragA(const float* __restrict__ p, int row0,
                                     int ld, int k, int laneM, int laneHi) {
  return *(const v2f*)(p + (size_t)(row0 + laneM) * ld + k + 2 * laneHi);
}

// Asynchronously stage a 256-row x 16-float (16 KB) tile from a row-major
// matrix (leading dim `ld` elements) into LDS, contiguous [row][16] layout.
// 256 threads x 4 x 16B each; tracked on ASYNCcnt (gfx1250 async path).
__device__ __forceinline__ void stage_tile_async(float* dst,
                                                 const float* __restrict__ src,
                                                 int ld) {
  const int t = threadIdx.x;
#pragma unroll
  for (int i = 0; i < 4; ++i) {
    const int f = i * 256 + t;        // 16-byte unit index, 0..1023
    const int row = f >> 2;           // 0..255
    const int col = (f & 3) << 2;     // float offset within the row
    const float* g = src + (size_t)row * ld + col;
    const uint32_t l = (uint32_t)(uintptr_t)(dst + f * 4);  // LDS byte offset
    asm volatile("global_load_async_to_lds_b128 %0, %1, off"
                 :: "v"(l), "v"(g) : "memory");
  }
}

__device__ __forceinline__ void wait_async0() {
  asm volatile("s_wait_asynccnt 0x0" ::: "memory");
}

// ---------------------------------------------------------------------------
// Kernel 1/5: Y[M,Nn] = X[M,K] @ W[Nn,K]^T + bias  (torch Linear semantics)
// Block = 8 waves, 128(M) x 64(N) macro-tile. W tile is shared by all waves:
// double-buffered async-staged in LDS. A fragments register-preloaded/chunk.
// ---------------------------------------------------------------------------
__global__ __launch_bounds__(256) void proj_abt_bias(
    const float* __restrict__ X, const float* __restrict__ W,
    const float* __restrict__ bias, float* __restrict__ Y,
    int Nn, int K) {
  __shared__ __align__(16) float wtile[2][64 * KC];  // [buf][n(64)][k(KC)]

  const int lane   = threadIdx.x & 31;
  const int wave   = threadIdx.x >> 5;
  const int laneM  = lane & 15;
  const int laneHi = lane >> 4;
  const int m0 = blockIdx.y * 128 + wave * 16;
  const int n0 = blockIdx.x * 64;
  const int nch = K / KC;

  stage_tile_async(wtile[0], W + (size_t)n0 * K, K);
  wait_async0();
  __syncthreads();

  v8f c0 = {}, c1 = {}, c2 = {}, c3 = {};
  for (int ch = 0; ch < nch; ++ch) {
    const int cur = ch & 1;
    if (ch + 1 < nch)  // uniform branch: EXEC stays full
      stage_tile_async(wtile[cur ^ 1], W + (size_t)n0 * K + (ch + 1) * KC, K);

    v2f afr[KC / 4];
#pragma unroll
    for (int s = 0; s < KC / 4; ++s)
      afr[s] = fragA(X, m0, K, ch * KC + 4 * s, laneM, laneHi);

    const float* tb = wtile[cur];
#pragma unroll
    for (int s = 0; s < KC / 4; ++s) {
      const int kk = 4 * s + 2 * laneHi;
      v2f b0 = *(const v2f*)&tb[(laneM     ) * KC + kk];
      v2f b1 = *(const v2f*)&tb[(laneM + 16) * KC + kk];
      v2f b2 = *(const v2f*)&tb[(laneM + 32) * KC + kk];
      v2f b3 = *(const v2f*)&tb[(laneM + 48) * KC + kk];
      c0 = wmma4(afr[s], b0, c0);
      c1 = wmma4(afr[s], b1, c1);
      c2 = wmma4(afr[s], b2, c2);
      c3 = wmma4(afr[s], b3, c3);
    }
    wait_async0();
    __syncthreads();
  }

  const float bv0 = bias[n0 + 0  + laneM];
  const float bv1 = bias[n0 + 16 + laneM];
  const float bv2 = bias[n0 + 32 + laneM];
  const float bv3 = bias[n0 + 48 + laneM];
#pragma unroll
  for (int r = 0; r < 8; ++r) {
    const int row = m0 + r + 8 * laneHi;  // C/D: v[r] holds M=r and M=r+8
    float* yp = Y + (size_t)row * Nn + n0 + laneM;
    yp[0]  = c0[r] + bv0;
    yp[16] = c1[r] + bv1;
    yp[32] = c2[r] + bv2;
    yp[48] = c3[r] + bv3;
  }
}

// ---------------------------------------------------------------------------
// Kernel 2/5: energy[z,qi,ki] = sum_d q[n,qi,h,d] * k[n,ki,h,d]
// Per (n,h): A(2048x64) @ B(2048x64)^T, both strided by EMBED. K=64 fits one
// staged tile: single async stage, then 16 fully-register-resident steps.
// ---------------------------------------------------------------------------
__global__ __launch_bounds__(256) void energy_qkt(
    const float* __restrict__ q, const float* __restrict__ kk,
    float* __restrict__ energy) {
  __shared__ __align__(16) float ktile[64 * HDIM];  // [ki(64)][d(64)]

  const int z = blockIdx.z;
  const int n = z >> 4;
  const int h = z & 15;
  const float* qb = q  + (size_t)n * SEQ * EMBED + h * HDIM;
  const float* kb = kk + (size_t)n * SEQ * EMBED + h * HDIM;
  float* eb = energy + (size_t)z * SEQ * SEQ;

  const int lane   = threadIdx.x & 31;
  const int wave   = threadIdx.x >> 5;
  const int laneM  = lane & 15;
  const int laneHi = lane >> 4;
  const int m0 = blockIdx.y * 128 + wave * 16;
  const int n0 = blockIdx.x * 64;

  stage_tile_async(ktile, kb + (size_t)n0 * EMBED, EMBED);

  v2f afr[HDIM / 4];
#pragma unroll
  for (int s = 0; s < HDIM / 4; ++s)
    afr[s] = fragA(qb, m0, EMBED, 4 * s, laneM, laneHi);

  wait_async0();
  __syncthreads();

  v8f c0 = {}, c1 = {}, c2 = {}, c3 = {};
#pragma unroll
  for (int s = 0; s < HDIM / 4; ++s) {
    const int kkk = 4 * s + 2 * laneHi;
    v2f b0 = *(const v2f*)&ktile[(laneM     ) * HDIM + kkk];
    v2f b1 = *(const v2f*)&ktile[(laneM + 16) * HDIM + kkk];
    v2f b2 = *(const v2f*)&ktile[(laneM + 32) * HDIM + kkk];
    v2f b3 = *(const v2f*)&ktile[(laneM + 48) * HDIM + kkk];
    c0 = wmma4(afr[s], b0, c0);
    c1 = wmma4(afr[s], b1, c1);
    c2 = wmma4(afr[s], b2, c2);
    c3 = wmma4(afr[s], b3, c3);
  }

#pragma unroll
  for (int r = 0; r < 8; ++r) {
    const int row = m0 + r + 8 * laneHi;
    float* ep = eb + (size_t)row * SEQ + n0 + laneM;
    ep[0]  = c0[r];
    ep[16] = c1[r];
    ep[32] = c2[r];
    ep[48] = c3[r];
  }
}

// ---------------------------------------------------------------------------
// Kernel 3/5: attention = softmax(energy / sqrt(HDIM)) along last axis.
// One 256-thread block per 2048-row; one read + one write of the 536MB tensor.
// ---------------------------------------------------------------------------
__global__ __launch_bounds__(256) void softmax_rows(
    const float* __restrict__ e, float* __restrict__ att) {
  const float scale = 0.125f;  // 1/sqrt(64)
  const size_t rid = blockIdx.x;
  const float* row = e + rid * SEQ;
  float* orow = att + rid * SEQ;
  const int t = threadIdx.x;

  __shared__ float sd[256];

  float r[8];
  float mx = -3.402823466e38f;
#pragma unroll
  for (int i = 0; i < 8; ++i) {
    r[i] = row[t + (i << 8)];
    mx = fmaxf(mx, r[i]);
  }
  sd[t] = mx;
  __syncthreads();
#pragma unroll
  for (int s = 128; s > 0; s >>= 1) {
    if (t < s) sd[t] = fmaxf(sd[t], sd[t + s]);
    __syncthreads();
  }
  const float m = sd[0] * scale;
  __syncthreads();

  float sum = 0.0f;
#pragma unroll
  for (int i = 0; i < 8; ++i) {
    r[i] = __expf(r[i] * scale - m);
    sum += r[i];
  }
  sd[t] = sum;
  __syncthreads();
#pragma unroll
  for (int s = 128; s > 0; s >>= 1) {
    if (t < s) sd[t] += sd[t + s];
    __syncthreads();
  }
  const float inv = 1.0f / sd[0];
#pragma unroll
  for (int i = 0; i < 8; ++i) orow[t + (i << 8)] = r[i] * inv;
}

// ---------------------------------------------------------------------------
// Kernel 4/5: attn_out[n,qi,h,d] = sum_l attention[z,qi,l] * v[n,l,h,d]
// Per (n,h): A(2048x2048) @ B(2048x64, ld=EMBED). v chunk (shared by all 8
// waves) double-buffered async-staged in LDS as [kk(64)][d(64)].
// ---------------------------------------------------------------------------
__global__ __launch_bounds__(256) void attn_pv(
    const float* __restrict__ att, const float* __restrict__ v,
    float* __restrict__ ws) {
  __shared__ __align__(16) float vtile[2][KC * 64];  // [buf][kk(64)][d(64)]

  const int z = blockIdx.z;
  const int n = z >> 4;
  const int h = z & 15;
  const float* ab = att + (size_t)z * SEQ * SEQ;
  const float* vb = v + (size_t)n * SEQ * EMBED + h * HDIM;

  const int lane   = threadIdx.x & 31;
  const int wave   = threadIdx.x >> 5;
  const int laneM  = lane & 15;
  const int laneHi = lane >> 4;
  const int m0 = blockIdx.y * 128 + wave * 16;
  const int nch = SEQ / KC;  // 32

  stage_tile_async(vtile[0], vb, EMBED);
  wait_async0();
  __syncthreads();

  v8f c0 = {}, c1 = {}, c2 = {}, c3 = {};
  for (int ch = 0; ch < nch; ++ch) {
    const int cur = ch & 1;
    if (ch + 1 < nch)
      stage_tile_async(vtile[cur ^ 1], vb + (size_t)(ch + 1) * KC * EMBED,
                       EMBED);

    v2f afr[KC / 4];
#pragma unroll
    for (int s = 0; s < KC / 4; ++s)
      afr[s] = fragA(ab, m0, SEQ, ch * KC + 4 * s, laneM, laneHi);

    const float* tb = vtile[cur];
#pragma unroll
    for (int s = 0; s < KC / 4; ++s) {
      const int kk = 4 * s + 2 * laneHi;
      v2f b0, b1, b2, b3;
      b0.x = tb[(kk    ) * 64 + laneM     ];
      b0.y = tb[(kk + 1) * 64 + laneM     ];
      b1.x = tb[(kk    ) * 64 + laneM + 16];
      b1.y = tb[(kk + 1) * 64 + laneM + 16];
      b2.x = tb[(kk    ) * 64 + laneM + 32];
      b2.y = tb[(kk + 1) * 64 + laneM + 32];
      b3.x = tb[(kk    ) * 64 + laneM + 48];
      b3.y = tb[(kk + 1) * 64 + laneM + 48];
      c0 = wmma4(afr[s], b0, c0);
      c1 = wmma4(afr[s], b1, c1);
      c2 = wmma4(afr[s], b2, c2);
      c3 = wmma4(afr[s], b3, c3);
    }
    wait_async0();
    __syncthreads();
  }

#pragma unroll
  for (int r = 0; r < 8; ++r) {
    const int row = m0 + r + 8 * laneHi;
    float* op = ws + ((size_t)n * SEQ + row) * EMBED + h * HDIM + laneM;
    op[0]  = c0[r];
    op[16] = c1[r];
    op[32] = c2[r];
    op[48] = c3[r];
  }
}

// ---------------------------------------------------------------------------
// Host-side launcher
// ---------------------------------------------------------------------------
extern "C" void kernel_launch(void* const* d_in, const int* in_sizes, int n_in,
                              void* d_out, int out_size, void* d_ws,
                              size_t ws_size, hipStream_t stream) {
  (void)in_sizes; (void)n_in; (void)out_size; (void)ws_size;
  const float* values = (const float*)d_in[0];
  const float* keys   = (const float*)d_in[1];
  const float* query  = (const float*)d_in[2];
  const float* Wq = (const float*)d_in[3];
  const float* bq = (const float*)d_in[4];
  const float* Wk = (const float*)d_in[5];
  const float* bk = (const float*)d_in[6];
  const float* Wv = (const float*)d_in[7];
  const float* bv = (const float*)d_in[8];
  const float* Wo = (const float*)d_in[9];
  const float* bo = (const float*)d_in[10];

  float* out = (float*)d_out;
  const size_t SZ_TOK = (size_t)NBAT * SEQ * EMBED;       // 4,194,304
  const size_t SZ_E   = (size_t)NBAT * HEADS * SEQ * SEQ; // 134,217,728
  float* o_out = out;                 // (N, L, E)
  float* o_q   = out + SZ_TOK;        // (N, L, H, D) == (N*L, E) row-major
  float* o_k   = out + 2 * SZ_TOK;
  float* o_v   = out + 3 * SZ_TOK;
  float* o_e   = out + 4 * SZ_TOK;    // (N, H, L, L)
  float* o_a   = o_e + SZ_E;          // (N, H, L, L)
  float* ws    = (float*)d_ws;        // attn_out staging (N, L, E) = 16 MB

  const dim3 blk(256);
  const dim3 gproj(EMBED / 64, (NBAT * SEQ) / 128);

  proj_abt_bias<<<gproj, blk, 0, stream>>>(query,  Wq, bq, o_q, EMBED, EMBED);
  proj_abt_bias<<<gproj, blk, 0, stream>>>(keys,   Wk, bk, o_k, EMBED, EMBED);
  proj_abt_bias<<<gproj, blk, 0, stream>>>(values, Wv, bv, o_v, EMBED, EMBED);

  energy_qkt<<<dim3(SEQ / 64, SEQ / 128, NBAT * HEADS), blk, 0, stream>>>(
      o_q, o_k, o_e);

  softmax_rows<<<dim3(NBAT * HEADS * SEQ), blk, 0, stream>>>(o_e, o_a);

  attn_pv<<<dim3(1, SEQ / 128, NBAT * HEADS), blk, 0, stream>>>(o_a, o_v, ws);

  proj_abt_bias<<<gproj, blk, 0, stream>>>(ws, Wo, bo, o_out, EMBED, EMBED);
}